// SelfAttention_81853486727949
// MI455X (gfx1250) — compile-verified
//
#include <hip/hip_runtime.h>
#include <math.h>

// ---------------------------------------------------------------------------
// CDNA5 (gfx1250) fp32 self-attention: QKV GEMM + RoPE + flash-attn + out GEMM
// All matrix math on V_WMMA_F32_16X16X4_F32 (wave32, fp32 in/out).
// ---------------------------------------------------------------------------

typedef __attribute__((ext_vector_type(2))) float v2f;
typedef __attribute__((ext_vector_type(8))) float v8f;

static __device__ __forceinline__ v8f wmma_f32(v2f a, v2f b, v8f c) {
  // 8 args: (neg_a, A, neg_b, B, c_mod, C, reuse_a, reuse_b)
  return __builtin_amdgcn_wmma_f32_16x16x4_f32(false, a, false, b, (short)0, c,
                                               false, false);
}

// ---------------------------------------------------------------------------
// Generic fp32 GEMM: C[M,N] = A[M,K] * B[K,N], row-major.
// WG = 256 threads (8 waves). WG tile 128x128, K-step 32.
// Wave w computes a 16x128 strip (8 accumulator tiles of 16x16).
// A tile is stored k-major (transposed) in LDS so A-fragment reads are
// consecutive-lane; B tile is natural k-major. Strides 132 floats:
// 132*4 = 528 B (16B-aligned rows) and 132 mod 64 = 4 (no bank conflicts).
// ---------------------------------------------------------------------------
__global__ __launch_bounds__(256) void gemm_f32_wmma(
    const float* __restrict__ A, const float* __restrict__ B,
    float* __restrict__ C, int M, int N, int K) {
  const int n0 = blockIdx.x * 128;
  const int m0 = blockIdx.y * 128;
  const int tid = threadIdx.x;
  const int lane = tid & 31;
  const int wave = tid >> 5;   // 0..7
  const int hlf = lane >> 4;   // 0 or 1 (half-wave)
  const int l16 = lane & 15;

  __shared__ float AsT[32][132];  // [k][m] (transposed A tile)
  __shared__ float Bs[32][132];   // [k][n]

  v8f acc[8] = {};

  for (int k0 = 0; k0 < K; k0 += 32) {
    // --- cooperative loads: A tile 128x32 (transpose into LDS) -------------
#pragma unroll
    for (int it = 0; it < 4; ++it) {
      int idx = tid + it * 256;  // 0..1023 float4s
      int row = idx >> 3;        // m: 0..127
      int c4 = idx & 7;          // k/4: 0..7
      float4 av = *(const float4*)(A + (size_t)(m0 + row) * K + k0 + c4 * 4);
      AsT[c4 * 4 + 0][row] = av.x;
      AsT[c4 * 4 + 1][row] = av.y;
      AsT[c4 * 4 + 2][row] = av.z;
      AsT[c4 * 4 + 3][row] = av.w;
    }
    // --- B tile 32x128 -----------------------------------------------------
#pragma unroll
    for (int it = 0; it < 4; ++it) {
      int idx = tid + it * 256;
      int row = idx >> 5;   // k: 0..31
      int c4 = idx & 31;    // n/4: 0..31
      float4 bv = *(const float4*)(B + (size_t)(k0 + row) * N + n0 + c4 * 4);
      *(float4*)&Bs[row][c4 * 4] = bv;
    }
    __syncthreads();

    // --- 8 WMMA K-steps of 4 ----------------------------------------------
#pragma unroll
    for (int ks = 0; ks < 8; ++ks) {
      const int kk = ks * 4 + hlf * 2;  // A layout: lanes16-31 hold K+2
      v2f a;
      a.x = AsT[kk + 0][wave * 16 + l16];
      a.y = AsT[kk + 1][wave * 16 + l16];
#pragma unroll
      for (int j = 0; j < 8; ++j) {
        v2f b;
        b.x = Bs[kk + 0][j * 16 + l16];
        b.y = Bs[kk + 1][j * 16 + l16];
        acc[j] = wmma_f32(a, b, acc[j]);
      }
    }
    __syncthreads();
  }

  // --- store C (C-layout: lane = col, vgpr v -> row v + 8*half) ------------
#pragma unroll
  for (int j = 0; j < 8; ++j) {
#pragma unroll
    for (int v = 0; v < 8; ++v) {
      int m = m0 + wave * 16 + v + hlf * 8;
      int n = n0 + j * 16 + l16;
      C[(size_t)m * N + n] = acc[j][v];
    }
  }
}

// ---------------------------------------------------------------------------
// RoPE, in place on the q and k thirds of the QKV buffer [B*T, 3072].
// One thread per (matrix, row, head, pair-index i in 0..31).
// q'[i]    = q[i]*cos - q[i+32]*sin ; q'[i+32] = q[i+32]*cos + q[i]*sin
// angle(t,i) = t * 10000^(-i/32)
// ---------------------------------------------------------------------------
__global__ __launch_bounds__(256) void rope_kernel(float* __restrict__ qkv,
                                                   int BT, int T) {
  int idx = blockIdx.x * blockDim.x + threadIdx.x;
  int total = 2 * BT * 16 * 32;
  if (idx >= total) return;
  int which = idx / (BT * 512);  // 0 = q, 1 = k
  int p = idx % (BT * 512);
  int r = p >> 9;          // row in [0, BT)
  int h = (p >> 5) & 15;   // head
  int i = p & 31;          // pair index
  int t = r % T;           // sequence position
  float* base = qkv + (size_t)r * 3072 + which * 1024 + h * 64;
  // inv_freq = exp(-i/32 * ln(10000))
  float inv = __expf(-(float)i * (9.210340371976184f / 32.0f));
  float ang = (float)t * inv;
  float s, c;
  sincosf(ang, &s, &c);
  float x1 = base[i];
  float x2 = base[i + 32];
  base[i] = x1 * c - x2 * s;
  base[i + 32] = x2 * c + x1 * s;
}

// ---------------------------------------------------------------------------
// Causal flash attention. Q,K,V read strided from QKV [B*T, 3072]:
//   q[b,h,t,d] = qkv[(b*T+t)*3072 +    0 + h*64 + d]
//   k[...]                          + 1024
//   v[...]                          + 2048
// Grid: (T/128, H, B), 256 threads (8 waves). Wave w owns q-rows w*16..+15.
// BLOCK_N = 64 K/V rows per step. Dh = 64. Online softmax; O normalized at
// the end and written to [B, T, H*64] (natural layout for the out-proj GEMM).
// ---------------------------------------------------------------------------
__global__ __launch_bounds__(256) void flash_attn_f32(
    const float* __restrict__ qkv, float* __restrict__ O, int T) {
  const int b = blockIdx.z;
  const int h = blockIdx.y;
  const int qm0 = blockIdx.x * 128;
  const int tid = threadIdx.x;
  const int lane = tid & 31;
  const int wave = tid >> 5;
  const int hlf = lane >> 4;
  const int l16 = lane & 15;

  __shared__ float QsT[64][132];     // [d][m]   (A-operand for S)
  __shared__ float KsT[64][68];      // [d][kv]  (B-operand for S = Q K^T)
  __shared__ float Vs[64][68];       // [kv][d]  (B-operand for P V)
  __shared__ float PsT[8][64][20];   // per-wave [kv][m] (A-operand for P V)

  const size_t ldq = 3072;
  const float* qbase = qkv + (size_t)b * T * ldq + h * 64;
  const float* kbase = qbase + 1024;
  const float* vbase = qbase + 2048;

  // ---- load Q tile (128 rows x 64) transposed into LDS ---------------------
#pragma unroll
  for (int it = 0; it < 8; ++it) {
    int idx = tid + it * 256;  // 0..2047 float4s
    int row = idx >> 4;        // 0..127
    int c4 = idx & 15;         // d/4
    float4 qv = *(const float4*)(qbase + (size_t)(qm0 + row) * ldq + c4 * 4);
    QsT[c4 * 4 + 0][row] = qv.x;
    QsT[c4 * 4 + 1][row] = qv.y;
    QsT[c4 * 4 + 2][row] = qv.z;
    QsT[c4 * 4 + 3][row] = qv.w;
  }

  v8f oacc[4] = {};
  float mrow[8], lrow[8];
#pragma unroll
  for (int v = 0; v < 8; ++v) {
    mrow[v] = -1e30f;
    lrow[v] = 0.0f;
  }
  const float scale = 0.125f;  // 1/sqrt(64)

  for (int kn0 = 0; kn0 < qm0 + 128; kn0 += 64) {
    __syncthreads();  // previous block's K/V no longer in use (also covers Q)
    // ---- load K (transposed) and V blocks, 64 rows each --------------------
#pragma unroll
    for (int it = 0; it < 4; ++it) {
      int idx = tid + it * 256;  // 0..1023 float4s
      int row = idx >> 4;        // kv row 0..63
      int c4 = idx & 15;
      float4 k4 = *(const float4*)(kbase + (size_t)(kn0 + row) * ldq + c4 * 4);
      KsT[c4 * 4 + 0][row] = k4.x;
      KsT[c4 * 4 + 1][row] = k4.y;
      KsT[c4 * 4 + 2][row] = k4.z;
      KsT[c4 * 4 + 3][row] = k4.w;
      float4 v4 = *(const float4*)(vbase + (size_t)(kn0 + row) * ldq + c4 * 4);
      *(float4*)&Vs[row][c4 * 4] = v4;
    }
    __syncthreads();

    // ---- S strip = Q(16x64) * K^T(64x64) for this wave ---------------------
    v8f sacc[4] = {};
#pragma unroll
    for (int ks = 0; ks < 16; ++ks) {
      const int kk = ks * 4 + hlf * 2;  // d index
      v2f a;
      a.x = QsT[kk + 0][wave * 16 + l16];
      a.y = QsT[kk + 1][wave * 16 + l16];
#pragma unroll
      for (int j = 0; j < 4; ++j) {
        v2f bf;
        bf.x = KsT[kk + 0][j * 16 + l16];
        bf.y = KsT[kk + 1][j * 16 + l16];
        sacc[j] = wmma_f32(a, bf, sacc[j]);
      }
    }

    // ---- scale + causal mask + online softmax ------------------------------
#pragma unroll
    for (int v = 0; v < 8; ++v) {
      const int mg = qm0 + wave * 16 + v + hlf * 8;  // global q row
      float mx = mrow[v];
#pragma unroll
      for (int j = 0; j < 4; ++j) {
        int ng = kn0 + j * 16 + l16;  // global k col
        float s = sacc[j][v] * scale;
        s = (ng <= mg) ? s : -1e30f;  // causal mask (exp underflows to 0)
        sacc[j][v] = s;
        mx = fmaxf(mx, s);
      }
#pragma unroll
      for (int off = 1; off <= 8; off <<= 1)  // reduce within 16-lane half
        mx = fmaxf(mx, __shfl_xor(mx, off, 32));
      float alpha = __expf(mrow[v] - mx);
      float sum = 0.0f;
#pragma unroll
      for (int j = 0; j < 4; ++j) {
        float p = __expf(sacc[j][v] - mx);
        sacc[j][v] = p;
        sum += p;
      }
#pragma unroll
      for (int off = 1; off <= 8; off <<= 1)
        sum += __shfl_xor(sum, off, 32);
      lrow[v] = lrow[v] * alpha + sum;
      mrow[v] = mx;
#pragma unroll
      for (int j = 0; j < 4; ++j) oacc[j][v] *= alpha;
    }

    // ---- P: C-layout -> A-layout via wave-private LDS ----------------------
#pragma unroll
    for (int j = 0; j < 4; ++j)
#pragma unroll
      for (int v = 0; v < 8; ++v)
        PsT[wave][j * 16 + l16][v + hlf * 8] = sacc[j][v];
    asm volatile("s_wait_dscnt 0" ::: "memory");  // wave-local LDS RAW

    // ---- O += P(16x64) * V(64x64) ------------------------------------------
#pragma unroll
    for (int ks = 0; ks < 16; ++ks) {
      const int kk = ks * 4 + hlf * 2;  // kv index
      v2f a;
      a.x = PsT[wave][kk + 0][l16];
      a.y = PsT[wave][kk + 1][l16];
#pragma unroll
      for (int j = 0; j < 4; ++j) {
        v2f bf;
        bf.x = Vs[kk + 0][j * 16 + l16];
        bf.y = Vs[kk + 1][j * 16 + l16];
        oacc[j] = wmma_f32(a, bf, oacc[j]);
      }
    }
  }

  // ---- normalize and store O[b, t, h*64 + d] -------------------------------
#pragma unroll
  for (int v = 0; v < 8; ++v) {
    const int m = qm0 + wave * 16 + v + hlf * 8;
    const float inv_l = 1.0f / lrow[v];
#pragma unroll
    for (int j = 0; j < 4; ++j)
      O[((size_t)b * T + m) * 1024 + h * 64 + j * 16 + l16] =
          oacc[j][v] * inv_l;
  }
}

// ---------------------------------------------------------------------------
// Launch: x[2,2048,1024] fp32, w_qkv[1024,3072], w_out[1024,1024] -> [2,2048,1024]
// Workspace: qkv (4096x3072 f32, 50.3 MB) + O (4096x1024 f32, 16.8 MB)
// ---------------------------------------------------------------------------
extern "C" void kernel_launch(void* const* d_in, const int* in_sizes, int n_in,
                              void* d_out, int out_size, void* d_ws,
                              size_t ws_size, hipStream_t stream) {
  (void)in_sizes; (void)n_in; (void)out_size; (void)ws_size;
  const float* x = (const float*)d_in[0];
  const float* w_qkv = (const float*)d_in[1];
  const float* w_out = (const float*)d_in[2];
  float* out = (float*)d_out;

  const int B = 2, T = 2048, C = 1024;
  const int M = B * T;  // 4096
  float* qkv = (float*)d_ws;
  float* Obuf = qkv + (size_t)M * 3 * C;

  dim3 blk(256);
  // 1) QKV projection: [4096,1024] x [1024,3072]
  gemm_f32_wmma<<<dim3(3 * C / 128, M / 128), blk, 0, stream>>>(
      x, w_qkv, qkv, M, 3 * C, C);
  // 2) RoPE in place on q,k thirds
  rope_kernel<<<(2 * M * 16 * 32) / 256, blk, 0, stream>>>(qkv, M, T);
  // 3) causal flash attention -> Obuf [B,T,C]
  flash_attn_f32<<<dim3(T / 128, 16, B), blk, 0, stream>>>(qkv, Obuf, T);
  // 4) output projection: [4096,1024] x [1024,1024]
  gemm_f32_wmma<<<dim3(C / 128, M / 128), blk, 0, stream>>>(
      Obuf, w_out, out, M, C, C);
}